// FNO2d_NIMRODglobal_3D_80771154969364
// MI455X (gfx1250) — compile-verified
//
#include <hip/hip_runtime.h>
#include <cmath>

#define NPTS 196608   // 64*64*48
#define PI2f 6.283185307179586f

typedef __attribute__((ext_vector_type(2))) float v2f;
typedef __attribute__((ext_vector_type(8))) float v8f;

__device__ __forceinline__ v8f wmma4(v2f a, v2f b, v8f c) {
  return __builtin_amdgcn_wmma_f32_16x16x4_f32(false, a, false, b, (short)0, c, false, false);
}
__device__ __forceinline__ float gelu_f(float x) {
  return 0.5f * x * (1.0f + erff(x * 0.7071067811865476f));
}
// group-g channel gather: xg blocks = {g, g, g+1, g+2} (mod 4)
__device__ __forceinline__ int srcchan(int c, int g) {
  int blk = c / 20;
  int m = (blk <= 1) ? g : ((g + blk - 1) & 3);
  return m * 20 + (c - blk * 20);
}

// ---------------- workspace layout (float offsets) ----------------
#define WS_WZ    0          // [48][16] fwd z twiddles
#define WS_WY    1024       // [128][32] fwd y
#define WS_WX    5120       // [128][32] fwd x
#define WS_WIX   9216       // [32][128] inv x
#define WS_WIY   13312      // [32][128] inv y
#define WS_WIZ   17408      // [16][48] inv z
#define WS_MEANH 18432
#define WS_RSTDH 18592
#define WS_MEANV 18752
#define WS_RSTDV 18792
#define WS_H     32768                    // (B,80,N)        125.8 MB
#define WS_FZ    (WS_H   + 31457280)      // (b,c,x,kz)[2y+p] 41.9 MB
#define WS_FZY   (WS_FZ  + 10485760)      // (b,c,ky,kz)[2x+p]10.5 MB
#define WS_FM    (WS_FZY + 2621440)       // (b,mode,reim,c)   2.6 MB
#define WS_GM    (WS_FM  + 655360)        // (b,o,ky,kz)[2kx+p]
#define WS_GZY   (WS_GM  + 163840)        // (b,o,x,kz)[2ky+p]
#define WS_GZ    (WS_GZY + 655360)        // (b,o,x,y)[2kz+p]
#define WS_V2    (WS_GZ  + 2621440)       // (B,20,N)         31.5 MB

// ---------------- twiddle generation ----------------
__global__ void k_twiddles(float* ws) {
  int idx = blockIdx.x * blockDim.x + threadIdx.x;
  if (idx < 768) {                 // WZ: out_re=cos, out_im=-sin (forward)
    int z = idx >> 4, nn = idx & 15, kz = nn >> 1;
    float ph = PI2f * (float)(kz * z) / 48.f;
    ws[WS_WZ + idx] = (nn & 1) ? -sinf(ph) : cosf(ph);
  } else if (idx < 1024) {
  } else if (idx < 5120) {         // WY forward: (Re+iIm)e^{-i ph}
    int j = idx - 1024; int k = j >> 5, nn = j & 31;
    int y = k >> 1, p = k & 1, ky = nn >> 1, r = nn & 1;
    float ph = PI2f * (float)(ky * y) / 64.f;
    float c = cosf(ph), s = sinf(ph);
    ws[WS_WY + j] = (p == 0) ? (r == 0 ? c : -s) : (r == 0 ? s : c);
  } else if (idx < 9216) {         // WX forward, freq map u -> u or u+48
    int j = idx - 5120; int k = j >> 5, nn = j & 31;
    int x = k >> 1, p = k & 1, u = nn >> 1, r = nn & 1;
    int f = (u < 8) ? u : u + 48;
    float ph = PI2f * (float)(f * x) / 64.f;
    float c = cosf(ph), s = sinf(ph);
    ws[WS_WX + j] = (p == 0) ? (r == 0 ? c : -s) : (r == 0 ? s : c);
  } else if (idx < 13312) {        // WIX inverse: (Re+iIm)e^{+i ph}/64
    int j = idx - 9216; int k = j >> 7, nn = j & 127;
    int u = k >> 1, p = k & 1, x = nn >> 1, r = nn & 1;
    int f = (u < 8) ? u : u + 48;
    float ph = PI2f * (float)(f * x) / 64.f;
    float c = cosf(ph) * (1.f / 64.f), s = sinf(ph) * (1.f / 64.f);
    ws[WS_WIX + j] = (p == 0) ? (r == 0 ? c : s) : (r == 0 ? -s : c);
  } else if (idx < 17408) {        // WIY inverse
    int j = idx - 13312; int k = j >> 7, nn = j & 127;
    int ky = k >> 1, p = k & 1, y = nn >> 1, r = nn & 1;
    float ph = PI2f * (float)(ky * y) / 64.f;
    float c = cosf(ph) * (1.f / 64.f), s = sinf(ph) * (1.f / 64.f);
    ws[WS_WIY + j] = (p == 0) ? (r == 0 ? c : s) : (r == 0 ? -s : c);
  } else if (idx < 18176) {        // WIZ irfft: (w_k/48)(Re cos - Im sin)
    int j = idx - 17408; int k = j / 48, z = j - 48 * k;
    int kz = k >> 1, p = k & 1;
    float ph = PI2f * (float)(kz * z) / 48.f;
    float w = ((kz == 0) ? 1.f : 2.f) / 48.f;
    ws[WS_WIZ + j] = (p == 0) ? w * cosf(ph) : -w * sinf(ph);
  }
}

// ---------------- lifting ----------------
__global__ void k_lift(const float* __restrict__ xin, const float* __restrict__ pw,
                       const float* __restrict__ pb, float* __restrict__ h) {
  int t = blockIdx.x * blockDim.x + threadIdx.x;      // 2*NPTS threads
  int b = t / NPTS, n = t - b * NPTS;
  int xi = n / 3072, yi = (n / 48) & 63, zi = n % 48;
  float gx = xi * (1.f / 63.f), gy = yi * (1.f / 63.f), gz = zi * (1.f / 47.f);
  const float* xp = xin + (size_t)t * 40;
  for (int g = 0; g < 4; ++g) {
    float in[13];
#pragma unroll
    for (int j = 0; j < 10; ++j) in[j] = xp[g * 10 + j];
    in[10] = gx; in[11] = gy; in[12] = gz;
    for (int o = 0; o < 20; ++o) {
      float a = pb[g * 20 + o];
#pragma unroll
      for (int j = 0; j < 13; ++j) a += in[j] * pw[(g * 13 + j) * 20 + o];
      h[((size_t)(b * 80 + g * 20 + o)) * NPTS + n] = a;
    }
  }
}

// ---------------- per-channel stats (mean / rstd over N) ----------------
__global__ void k_stats(const float* __restrict__ src, float* __restrict__ meanA,
                        float* __restrict__ rstdA, int c0, int nc, int Ctot) {
  int b = blockIdx.x / nc, c = c0 + blockIdx.x % nc;
  const float* p = src + ((size_t)(b * Ctot + c)) * NPTS;
  float s = 0.f, ss = 0.f;
  for (int n = threadIdx.x; n < NPTS; n += blockDim.x) { float v = p[n]; s += v; ss += v * v; }
  __shared__ float sh0[256]; __shared__ float sh1[256];
  sh0[threadIdx.x] = s; sh1[threadIdx.x] = ss; __syncthreads();
  for (int off = 128; off > 0; off >>= 1) {
    if ((int)threadIdx.x < off) { sh0[threadIdx.x] += sh0[threadIdx.x + off]; sh1[threadIdx.x] += sh1[threadIdx.x + off]; }
    __syncthreads();
  }
  if (threadIdx.x == 0) {
    float m = sh0[0] / (float)NPTS;
    float var = sh1[0] / (float)NPTS - m * m;
    meanA[b * Ctot + c] = m;
    rstdA[b * Ctot + c] = rsqrtf(var + 1e-5f);
  }
}

// ---------------- forward DFT stage Z (norm fused into A) ----------------
__global__ void k_dftz(const float* __restrict__ h, const float* __restrict__ mean,
                       const float* __restrict__ rstd, const float* __restrict__ Wz,
                       float* __restrict__ Fz, int grp) {
  int wv = blockIdx.x * 8 + (threadIdx.x >> 5);
  int lane = threadIdx.x & 31, laneM = lane & 15, laneH = lane >> 4;
  int row0 = wv << 4;
  int row = row0 + laneM;                       // (b,c,x,y)
  int b = row / (80 * 4096), rem = row - b * 80 * 4096;
  int c = rem / 4096, xy = rem - c * 4096;
  int sc = srcchan(c, grp);
  const float* Ap = h + ((size_t)(b * 80 + sc)) * NPTS + (size_t)xy * 48;
  float mu = mean[b * 80 + sc], rs = rstd[b * 80 + sc];
  v8f acc = {};
  for (int k0 = 0; k0 < 48; k0 += 4) {
    int kk = k0 + 2 * laneH;
    v2f a, bb;
    a.x = (Ap[kk] - mu) * rs;
    a.y = (Ap[kk + 1] - mu) * rs;
    bb.x = Wz[kk * 16 + laneM];
    bb.y = Wz[(kk + 1) * 16 + laneM];
    acc = wmma4(a, bb, acc);
  }
  int bcx = row0 >> 6, y0 = row0 & 63;
  int kz = laneM >> 1, p = laneM & 1;
  float* Op = Fz + (size_t)bcx * 1024 + kz * 128 + p;
#pragma unroll
  for (int j = 0; j < 8; ++j) Op[2 * (y0 + j + 8 * laneH)] = acc[j];
}

// ---------------- forward DFT stage Y ----------------
__global__ void k_dfty(const float* __restrict__ Fz, const float* __restrict__ Wy,
                       float* __restrict__ Fzy) {
  int wv = blockIdx.x * 8 + (threadIdx.x >> 5);
  int lane = threadIdx.x & 31, laneM = lane & 15, laneH = lane >> 4;
  int row0 = (wv >> 1) << 4, n0 = (wv & 1) << 4;
  const float* Ap = Fz + (size_t)(row0 + laneM) * 128;
  v8f acc = {};
  for (int k0 = 0; k0 < 128; k0 += 4) {
    int kk = k0 + 2 * laneH;
    v2f a, bb;
    a.x = Ap[kk]; a.y = Ap[kk + 1];
    bb.x = Wy[kk * 32 + n0 + laneM];
    bb.y = Wy[(kk + 1) * 32 + n0 + laneM];
    acc = wmma4(a, bb, acc);
  }
  int n = n0 + laneM, ky = n >> 1, r = n & 1;
#pragma unroll
  for (int j = 0; j < 8; ++j) {
    int q = row0 + j + 8 * laneH;                // (b,c,x,kz)
    int bc = q >> 9, x = (q >> 3) & 63, kz = q & 7;
    Fzy[(size_t)((bc * 16 + ky) * 8 + kz) * 128 + 2 * x + r] = acc[j];
  }
}

// ---------------- forward DFT stage X ----------------
__global__ void k_dftx(const float* __restrict__ Fzy, const float* __restrict__ Wx,
                       float* __restrict__ Fm) {
  int wv = blockIdx.x * 8 + (threadIdx.x >> 5);
  int lane = threadIdx.x & 31, laneM = lane & 15, laneH = lane >> 4;
  int row0 = (wv >> 1) << 4, n0 = (wv & 1) << 4;
  const float* Ap = Fzy + (size_t)(row0 + laneM) * 128;
  v8f acc = {};
  for (int k0 = 0; k0 < 128; k0 += 4) {
    int kk = k0 + 2 * laneH;
    v2f a, bb;
    a.x = Ap[kk]; a.y = Ap[kk + 1];
    bb.x = Wx[kk * 32 + n0 + laneM];
    bb.y = Wx[(kk + 1) * 32 + n0 + laneM];
    acc = wmma4(a, bb, acc);
  }
  int n = n0 + laneM, u = n >> 1, r = n & 1;
#pragma unroll
  for (int j = 0; j < 8; ++j) {
    int q = row0 + j + 8 * laneH;                // (b,c,ky,kz)
    int bc = q >> 7, ky = (q >> 3) & 15, kz = q & 7;
    int b = bc / 80, c = bc - 80 * b;
    int mode = u * 128 + ky * 8 + kz;
    Fm[(size_t)((b * 2048 + mode) * 2 + r) * 80 + c] = acc[j];
  }
}

// ---------------- per-mode spectral multiply (per-mode 80->20 complex) ----------------
__global__ void k_spectral(const float* __restrict__ Fm, const float* __restrict__ spec,
                           float* __restrict__ Gm, int l, int grp) {
  int t = blockIdx.x * blockDim.x + threadIdx.x;       // 2*20*2048
  int b = t / 40960, rem = t - b * 40960;
  int o = rem >> 11, mode = rem & 2047;
  int kx = mode >> 7, ky = (mode >> 3) & 15, kz = mode & 7;
  int quad = (kx >= 8 ? 1 : 0) + (ky >= 8 ? 2 : 0);
  int m1 = kx & 7, m2 = ky & 7, m3 = kz;
  int qb = ((l * 4 + grp) * 4 + quad) * 80;
  const float* X = Fm + (size_t)(b * 2048 + mode) * 160;         // re[80], im[80]
  const float* Wp = spec + ((size_t)(qb * 20 + o)) * 1024 + (m1 * 64 + m2 * 8 + m3) * 2;
  float ar = 0.f, ai = 0.f;
  for (int c = 0; c < 80; ++c) {
    float xr = X[c], xim = X[80 + c];
    float wr = Wp[0], wi = Wp[1];
    ar += xr * wr - xim * wi;
    ai += xr * wi + xim * wr;
    Wp += 20480;                                      // next c (stride 20*8*8*8*2)
  }
  size_t q = (size_t)((b * 20 + o) * 128 + ky * 8 + kz) * 32;
  Gm[q + 2 * kx] = ar;
  Gm[q + 2 * kx + 1] = ai;
}

// ---------------- inverse DFT stage X ----------------
__global__ void k_idftx(const float* __restrict__ Gm, const float* __restrict__ Wix,
                        float* __restrict__ Gzy) {
  int wv = blockIdx.x * 8 + (threadIdx.x >> 5);
  int lane = threadIdx.x & 31, laneM = lane & 15, laneH = lane >> 4;
  int row0 = (wv >> 3) << 4, n0 = (wv & 7) << 4;
  const float* Ap = Gm + (size_t)(row0 + laneM) * 32;
  v8f acc = {};
  for (int k0 = 0; k0 < 32; k0 += 4) {
    int kk = k0 + 2 * laneH;
    v2f a, bb;
    a.x = Ap[kk]; a.y = Ap[kk + 1];
    bb.x = Wix[kk * 128 + n0 + laneM];
    bb.y = Wix[(kk + 1) * 128 + n0 + laneM];
    acc = wmma4(a, bb, acc);
  }
  int n = n0 + laneM, x = n >> 1, r = n & 1;
#pragma unroll
  for (int j = 0; j < 8; ++j) {
    int q = row0 + j + 8 * laneH;                 // (b,o,ky,kz)
    int bo = q >> 7, ky = (q >> 3) & 15, kz = q & 7;
    Gzy[(size_t)((bo * 64 + x) * 8 + kz) * 32 + 2 * ky + r] = acc[j];
  }
}

// ---------------- inverse DFT stage Y ----------------
__global__ void k_idfty(const float* __restrict__ Gzy, const float* __restrict__ Wiy,
                        float* __restrict__ Gz) {
  int wv = blockIdx.x * 8 + (threadIdx.x >> 5);
  int lane = threadIdx.x & 31, laneM = lane & 15, laneH = lane >> 4;
  int row0 = (wv >> 3) << 4, n0 = (wv & 7) << 4;
  const float* Ap = Gzy + (size_t)(row0 + laneM) * 32;
  v8f acc = {};
  for (int k0 = 0; k0 < 32; k0 += 4) {
    int kk = k0 + 2 * laneH;
    v2f a, bb;
    a.x = Ap[kk]; a.y = Ap[kk + 1];
    bb.x = Wiy[kk * 128 + n0 + laneM];
    bb.y = Wiy[(kk + 1) * 128 + n0 + laneM];
    acc = wmma4(a, bb, acc);
  }
  int n = n0 + laneM, y = n >> 1, r = n & 1;
#pragma unroll
  for (int j = 0; j < 8; ++j) {
    int q = row0 + j + 8 * laneH;                 // (b,o,x,kz)
    int bo = q >> 9, x = (q >> 3) & 63, kz = q & 7;
    Gz[(size_t)((bo * 64 + x) * 64 + y) * 16 + 2 * kz + r] = acc[j];
  }
}

// ---------------- inverse DFT stage Z (to spatial v2) ----------------
__global__ void k_idftz(const float* __restrict__ Gz, const float* __restrict__ Wiz,
                        float* __restrict__ v2) {
  int wv = blockIdx.x * 8 + (threadIdx.x >> 5);
  int lane = threadIdx.x & 31, laneM = lane & 15, laneH = lane >> 4;
  int row0 = (wv / 3) << 4, n0 = (wv % 3) << 4;
  const float* Ap = Gz + (size_t)(row0 + laneM) * 16;
  v8f acc = {};
  for (int k0 = 0; k0 < 16; k0 += 4) {
    int kk = k0 + 2 * laneH;
    v2f a, bb;
    a.x = Ap[kk]; a.y = Ap[kk + 1];
    bb.x = Wiz[kk * 48 + n0 + laneM];
    bb.y = Wiz[(kk + 1) * 48 + n0 + laneM];
    acc = wmma4(a, bb, acc);
  }
  int z = n0 + laneM;
#pragma unroll
  for (int j = 0; j < 8; ++j) {
    int q = row0 + j + 8 * laneH;                 // (b,o,x,y)
    int bo = q >> 12, xy = q & 4095;
    v2[(size_t)bo * NPTS + xy * 48 + z] = acc[j];
  }
}

// ---------------- fused norm2 + MLP + residual conv + h update ----------------
__global__ void k_mlp_res(float* __restrict__ h, const float* __restrict__ v2,
                          const float* __restrict__ mean2, const float* __restrict__ rstd2,
                          const float* __restrict__ w1, const float* __restrict__ b1,
                          const float* __restrict__ w2, const float* __restrict__ b2,
                          const float* __restrict__ ww, const float* __restrict__ wb,
                          int l, int grp) {
  __shared__ float lt[8][16][21];
  int warp = threadIdx.x >> 5;
  int wv = blockIdx.x * 8 + warp;
  int lane = threadIdx.x & 31, laneM = lane & 15, laneH = lane >> 4;
  int row0 = wv << 4;
  int b = row0 / NPTS, n0 = row0 - b * NPTS;
  int n = n0 + laneM;
  int li = l * 4 + grp;
  const float* w1p = w1 + li * 400;  const float* b1p = b1 + li * 20;
  const float* w2p = w2 + li * 400;  const float* b2p = b2 + li * 20;
  const float* wwp = ww + li * 1600; const float* wbp = wb + li * 20;
  int o0 = laneM, o1 = 16 + laneM;
  // GEMM1: gelu(norm(v2) @ W1 + b1)
  v8f a0 = {}; v8f a1 = {};
  for (int k0 = 0; k0 < 20; k0 += 4) {
    int kk = k0 + 2 * laneH;
    int ci = b * 20 + kk;
    v2f a, bb0, bb1;
    a.x = (v2[(size_t)ci * NPTS + n] - mean2[ci]) * rstd2[ci];
    a.y = (v2[(size_t)(ci + 1) * NPTS + n] - mean2[ci + 1]) * rstd2[ci + 1];
    bb0.x = w1p[o0 * 20 + kk];  bb0.y = w1p[o0 * 20 + kk + 1];
    bb1.x = (o1 < 20) ? w1p[o1 * 20 + kk] : 0.f;
    bb1.y = (o1 < 20) ? w1p[o1 * 20 + kk + 1] : 0.f;
    a0 = wmma4(a, bb0, a0); a1 = wmma4(a, bb1, a1);
  }
  float bi0 = b1p[o0], bi1 = (o1 < 20) ? b1p[o1] : 0.f;
#pragma unroll
  for (int j = 0; j < 8; ++j) {
    int m = j + 8 * laneH;
    lt[warp][m][o0] = gelu_f(a0[j] + bi0);
    if (o1 < 20) lt[warp][m][o1] = gelu_f(a1[j] + bi1);
  }
  __syncthreads();
  // GEMM2 (t @ W2) then residual (gather(h) @ Ww^T) accumulated together
  v8f u0 = {}; v8f u1 = {};
  for (int k0 = 0; k0 < 20; k0 += 4) {
    int kk = k0 + 2 * laneH;
    v2f a, bb0, bb1;
    a.x = lt[warp][laneM][kk]; a.y = lt[warp][laneM][kk + 1];
    bb0.x = w2p[o0 * 20 + kk];  bb0.y = w2p[o0 * 20 + kk + 1];
    bb1.x = (o1 < 20) ? w2p[o1 * 20 + kk] : 0.f;
    bb1.y = (o1 < 20) ? w2p[o1 * 20 + kk + 1] : 0.f;
    u0 = wmma4(a, bb0, u0); u1 = wmma4(a, bb1, u1);
  }
  for (int k0 = 0; k0 < 80; k0 += 4) {
    int kk = k0 + 2 * laneH;
    int s0 = srcchan(kk, grp), s1 = srcchan(kk + 1, grp);
    v2f a, bb0, bb1;
    a.x = h[((size_t)(b * 80 + s0)) * NPTS + n];
    a.y = h[((size_t)(b * 80 + s1)) * NPTS + n];
    bb0.x = wwp[o0 * 80 + kk];  bb0.y = wwp[o0 * 80 + kk + 1];
    bb1.x = (o1 < 20) ? wwp[o1 * 80 + kk] : 0.f;
    bb1.y = (o1 < 20) ? wwp[o1 * 80 + kk + 1] : 0.f;
    u0 = wmma4(a, bb0, u0); u1 = wmma4(a, bb1, u1);
  }
  float f0 = b2p[o0] + wbp[o0];
  float f1 = (o1 < 20) ? (b2p[o1] + wbp[o1]) : 0.f;
#pragma unroll
  for (int j = 0; j < 8; ++j) {
    int nn = n0 + j + 8 * laneH;
    h[((size_t)(b * 80 + grp * 20 + o0)) * NPTS + nn] = u0[j] + f0;
    if (o1 < 20) h[((size_t)(b * 80 + grp * 20 + o1)) * NPTS + nn] = u1[j] + f1;
  }
}

// ---------------- elementwise gelu on h ----------------
__global__ void k_gelu(float* __restrict__ h) {
  int t = blockIdx.x * blockDim.x + threadIdx.x;
  h[t] = gelu_f(h[t]);
}

// ---------------- fused projection: gelu(h@Q1+b1)@Q2+b2 -> out (transposed) ----------------
__global__ void k_proj(const float* __restrict__ h, const float* __restrict__ qw1,
                       const float* __restrict__ qb1, const float* __restrict__ qw2,
                       const float* __restrict__ qb2, float* __restrict__ out, int grp) {
  __shared__ float lt[8][16][81];
  int warp = threadIdx.x >> 5;
  int wv = blockIdx.x * 8 + warp;
  int lane = threadIdx.x & 31, laneM = lane & 15, laneH = lane >> 4;
  int row0 = wv << 4;
  int b = row0 / NPTS, n0 = row0 - b * NPTS;
  int n = n0 + laneM;
  const float* w1p = qw1 + grp * 1600;  const float* b1p = qb1 + grp * 80;
  const float* w2p = qw2 + grp * 1600;  const float* b2p = qb2 + grp * 20;
  v8f z = {};
  v8f acc[5]; for (int t5 = 0; t5 < 5; ++t5) acc[t5] = z;
  for (int k0 = 0; k0 < 20; k0 += 4) {
    int kk = k0 + 2 * laneH;
    v2f a;
    a.x = h[((size_t)(b * 80 + grp * 20 + kk)) * NPTS + n];
    a.y = h[((size_t)(b * 80 + grp * 20 + kk + 1)) * NPTS + n];
    for (int ct = 0; ct < 5; ++ct) {
      int o = ct * 16 + laneM;
      v2f bb;
      bb.x = w1p[o * 20 + kk]; bb.y = w1p[o * 20 + kk + 1];
      acc[ct] = wmma4(a, bb, acc[ct]);
    }
  }
  for (int ct = 0; ct < 5; ++ct) {
    int o = ct * 16 + laneM;
    float bi = b1p[o];
#pragma unroll
    for (int j = 0; j < 8; ++j) lt[warp][j + 8 * laneH][o] = gelu_f(acc[ct][j] + bi);
  }
  __syncthreads();
  int o0 = laneM, o1 = 16 + laneM;
  v8f u0 = z, u1 = z;
  for (int k0 = 0; k0 < 80; k0 += 4) {
    int kk = k0 + 2 * laneH;
    v2f a, bb0, bb1;
    a.x = lt[warp][laneM][kk]; a.y = lt[warp][laneM][kk + 1];
    bb0.x = w2p[o0 * 80 + kk];  bb0.y = w2p[o0 * 80 + kk + 1];
    bb1.x = (o1 < 20) ? w2p[o1 * 80 + kk] : 0.f;
    bb1.y = (o1 < 20) ? w2p[o1 * 80 + kk + 1] : 0.f;
    u0 = wmma4(a, bb0, u0); u1 = wmma4(a, bb1, u1);
  }
#pragma unroll
  for (int j = 0; j < 8; ++j) {
    int nn = n0 + j + 8 * laneH;
    out[((size_t)(b * NPTS + nn)) * 80 + grp * 20 + o0] = u0[j] + b2p[o0];
    if (o1 < 20) out[((size_t)(b * NPTS + nn)) * 80 + grp * 20 + o1] = u1[j] + b2p[o1];
  }
}

// ---------------- host launcher ----------------
extern "C" void kernel_launch(void* const* d_in, const int* in_sizes, int n_in,
                              void* d_out, int out_size, void* d_ws, size_t ws_size,
                              hipStream_t stream) {
  const float* x_in = (const float*)d_in[0];
  const float* p_w  = (const float*)d_in[1];
  const float* p_b  = (const float*)d_in[2];
  const float* spw  = (const float*)d_in[3];
  const float* m_w1 = (const float*)d_in[4];
  const float* m_b1 = (const float*)d_in[5];
  const float* m_w2 = (const float*)d_in[6];
  const float* m_b2 = (const float*)d_in[7];
  const float* w_w  = (const float*)d_in[8];
  const float* w_b  = (const float*)d_in[9];
  const float* q_w1 = (const float*)d_in[10];
  const float* q_b1 = (const float*)d_in[11];
  const float* q_w2 = (const float*)d_in[12];
  const float* q_b2 = (const float*)d_in[13];
  float* ws = (float*)d_ws;
  float* WZ = ws + WS_WZ;   float* WY = ws + WS_WY;   float* WX = ws + WS_WX;
  float* WIX = ws + WS_WIX; float* WIY = ws + WS_WIY; float* WIZ = ws + WS_WIZ;
  float* MH = ws + WS_MEANH; float* RH = ws + WS_RSTDH;
  float* MV = ws + WS_MEANV; float* RV = ws + WS_RSTDV;
  float* H  = ws + WS_H;   float* FZ = ws + WS_FZ;   float* FZY = ws + WS_FZY;
  float* FM = ws + WS_FM;  float* GM = ws + WS_GM;   float* GZY = ws + WS_GZY;
  float* GZ = ws + WS_GZ;  float* V2 = ws + WS_V2;

  k_twiddles<<<71, 256, 0, stream>>>(ws);
  k_lift<<<1536, 256, 0, stream>>>(x_in, p_w, p_b, H);
  k_stats<<<160, 256, 0, stream>>>(H, MH, RH, 0, 80, 80);

  for (int l = 0; l < 4; ++l) {
    for (int g = 0; g < 4; ++g) {
      k_dftz<<<5120, 256, 0, stream>>>(H, MH, RH, WZ, FZ, g);
      k_dfty<<<1280, 256, 0, stream>>>(FZ, WY, FZY);
      k_dftx<<<320, 256, 0, stream>>>(FZY, WX, FM);
      k_spectral<<<320, 256, 0, stream>>>(FM, spw, GM, l, g);
      k_idftx<<<320, 256, 0, stream>>>(GM, WIX, GZY);
      k_idfty<<<1280, 256, 0, stream>>>(GZY, WIY, GZ);
      k_idftz<<<3840, 256, 0, stream>>>(GZ, WIZ, V2);
      k_stats<<<40, 256, 0, stream>>>(V2, MV, RV, 0, 20, 20);
      k_mlp_res<<<3072, 256, 0, stream>>>(H, V2, MV, RV, m_w1, m_b1, m_w2, m_b2,
                                          w_w, w_b, l, g);
      if (!(l == 3 && g == 3))  // incremental stats for the rewritten block
        k_stats<<<40, 256, 0, stream>>>(H, MH, RH, 20 * g, 20, 80);
    }
    if (l < 3) {
      k_gelu<<<122880, 256, 0, stream>>>(H);
      k_stats<<<160, 256, 0, stream>>>(H, MH, RH, 0, 80, 80);
    }
  }
  for (int g = 0; g < 4; ++g)
    k_proj<<<3072, 256, 0, stream>>>(H, q_w1, q_b1, q_w2, q_b2, (float*)d_out, g);
  (void)in_sizes; (void)n_in; (void)out_size; (void)ws_size;
}